// Dyna_81853486727658
// MI455X (gfx1250) — compile-verified
//
#include <hip/hip_runtime.h>

#define N_S   8      // batch
#define C_TOT 64     // channels
#define GRP   8      // groups
#define CGRP  8      // channels per group
#define H     256
#define W     256
#define KK    9      // 3x3 taps
#define FILT_N 72    // GRP * KK
#define PW    (W + 2)
#define ROWS  8

typedef __attribute__((ext_vector_type(2))) float v2f;
typedef __attribute__((ext_vector_type(8))) float v8f;

// ---------------------------------------------------------------------------
// Kernel A: global average pool. One block per (n,c) plane (512 planes).
// float4 loads -> 256-thread LDS tree reduction -> mean.
// ---------------------------------------------------------------------------
__global__ void __launch_bounds__(256)
gap_kernel(const float* __restrict__ x, float* __restrict__ gap) {
    const int plane = blockIdx.x;                      // n*64 + c
    const float4* p = (const float4*)(x + (size_t)plane * (H * W));
    float s = 0.f;
    #pragma unroll 4
    for (int i = threadIdx.x; i < (H * W) / 4; i += 256) {
        float4 v = p[i];
        s += (v.x + v.y) + (v.z + v.w);
    }
    __shared__ float red[256];
    red[threadIdx.x] = s;
    __syncthreads();
    for (int off = 128; off > 0; off >>= 1) {
        if ((int)threadIdx.x < off) red[threadIdx.x] += red[threadIdx.x + off];
        __syncthreads();
    }
    if (threadIdx.x == 0) gap[plane] = red[0] * (1.0f / (H * W));
}

// ---------------------------------------------------------------------------
// Kernel B: filt = tanh(gap @ conv_w^T) via V_WMMA_F32_16X16X4_F32.
// gap: (8,64)  conv_w: (72,64)  filt: (8,72).
// Operands are zero-padded into LDS first (sA: 16x64, sB: 80x64) so the
// 16-step K loop issues unconditional ds_load_b64 fragment reads -> v_wmma
// with no EXEC-mask branching.
// 5 wave32s; wave t computes output columns [16t, 16t+16).
// A 16x4 f32 frag: lanes 0-15 hold (K=k0,k0+1), lanes 16-31 hold (k0+2,k0+3).
// D 16x16 f32 frag: lanes 0-15 VGPR v = row M=v (our 8 valid samples).
// ---------------------------------------------------------------------------
__global__ void __launch_bounds__(160)
filt_kernel(const float* __restrict__ gap,
            const float* __restrict__ cw,
            float* __restrict__ filt) {
    __shared__ float sA[16 * 64];   // gap, rows 8..15 zero
    __shared__ float sB[80 * 64];   // sB[o][k] = conv_w[o][k], rows 72..79 zero

    const int tid = threadIdx.x;
    for (int f = tid; f < 16 * 64; f += 160)
        sA[f] = (f < N_S * 64) ? gap[f] : 0.f;
    for (int f = tid; f < 80 * 64; f += 160)
        sB[f] = (f < FILT_N * 64) ? cw[f] : 0.f;
    __syncthreads();

    const int wave = tid >> 5;             // 0..4 -> N tile
    const int lane = tid & 31;
    const int half = lane >> 4;            // which K half-pair this lane holds
    const int l16  = lane & 15;
    const int ncol = wave * 16 + l16;      // output column o (0..79)

    const float* aRow = &sA[l16 * 64];
    const float* bRow = &sB[ncol * 64];

    v8f acc = {};
    #pragma unroll
    for (int k0 = 0; k0 < 64; k0 += 4) {
        const int ka = k0 + half * 2;      // even -> 8B-aligned ds_load_b64
        v2f a, b;
        a[0] = aRow[ka];
        a[1] = aRow[ka + 1];
        b[0] = bRow[ka];
        b[1] = bRow[ka + 1];
        acc = __builtin_amdgcn_wmma_f32_16x16x4_f32(
                  /*neg_a=*/false, a, /*neg_b=*/false, b,
                  /*c_mod=*/(short)0, acc, /*reuse_a=*/false, /*reuse_b=*/false);
    }
    if (half == 0 && ncol < FILT_N) {
        #pragma unroll
        for (int v = 0; v < 8; ++v)        // M=v for lanes 0-15
            filt[v * FILT_N + ncol] = tanhf(acc[v]);
    }
}

// ---------------------------------------------------------------------------
// Kernel C: dynamic 3x3 conv (reflect pad) + residual.
// Block: 256 threads = one 8-row strip of one (n,c) plane.
// 10x258 reflect-padded halo staged in LDS; 9 weights broadcast in registers.
// ---------------------------------------------------------------------------
__global__ void __launch_bounds__(256)
dynaconv_kernel(const float* __restrict__ x,
                const float* __restrict__ filt,
                float* __restrict__ out) {
    const int tile = blockIdx.x;           // 0..31 row strips
    const int c    = blockIdx.y;           // 0..63
    const int n    = blockIdx.z;           // 0..7
    const int g    = c >> 3;               // group = c / CGRP
    const int h0   = tile * ROWS;

    __shared__ float lds[(ROWS + 2) * PW];

    float wt[KK];
    #pragma unroll
    for (int i = 0; i < KK; ++i) wt[i] = filt[n * FILT_N + g * KK + i];

    const float* plane = x + ((size_t)(n * C_TOT + c)) * (H * W);

    for (int f = threadIdx.x; f < (ROWS + 2) * PW; f += 256) {
        int r   = f / PW;
        int col = f - r * PW;
        int hh  = h0 + r - 1;
        hh = (hh < 0) ? -hh : ((hh >= H) ? (2 * H - 2 - hh) : hh);
        int ww  = col - 1;
        ww = (ww < 0) ? -ww : ((ww >= W) ? (2 * W - 2 - ww) : ww);
        lds[f] = plane[hh * W + ww];
    }
    __syncthreads();

    const int wcol = threadIdx.x;          // output column 0..255
    float* oplane = out + ((size_t)(n * C_TOT + c)) * (H * W);
    #pragma unroll
    for (int r = 0; r < ROWS; ++r) {
        const float* row0 = &lds[r * PW + wcol];
        float acc = row0[0]        * wt[0] + row0[1]        * wt[1] + row0[2]        * wt[2]
                  + row0[PW]       * wt[3] + row0[PW + 1]   * wt[4] + row0[PW + 2]   * wt[5]
                  + row0[2 * PW]   * wt[6] + row0[2 * PW+1] * wt[7] + row0[2 * PW+2] * wt[8];
        acc += row0[PW + 1];               // residual (center x)
        oplane[(h0 + r) * W + wcol] = acc;
    }
}

// ---------------------------------------------------------------------------
extern "C" void kernel_launch(void* const* d_in, const int* in_sizes, int n_in,
                              void* d_out, int out_size, void* d_ws, size_t ws_size,
                              hipStream_t stream) {
    const float* x  = (const float*)d_in[0];   // (8,64,256,256)
    const float* cw = (const float*)d_in[1];   // (72,64)
    float* out  = (float*)d_out;               // (8,64,256,256)
    float* gap  = (float*)d_ws;                // 512 floats
    float* filt = gap + 512;                   // 576 floats

    gap_kernel<<<dim3(N_S * C_TOT), dim3(256), 0, stream>>>(x, gap);
    filt_kernel<<<dim3(1), dim3(160), 0, stream>>>(gap, cw, filt);
    dim3 grid(H / ROWS, C_TOT, N_S);
    dynaconv_kernel<<<grid, dim3(256), 0, stream>>>(x, filt, out);
}